// AdditiveAttention_44564580663638
// MI455X (gfx1250) — compile-verified
//
#include <hip/hip_runtime.h>
#include <hip/hip_bf16.h>
#include <math.h>
#include <stdint.h>

typedef __attribute__((ext_vector_type(2))) float v2f;
typedef __attribute__((ext_vector_type(4))) float v4f;
typedef __attribute__((ext_vector_type(8))) float v8f;
typedef __attribute__((ext_vector_type(4))) unsigned int u32x4;
typedef __attribute__((ext_vector_type(8))) int i32x8;
typedef __attribute__((ext_vector_type(4))) int i32x4;

// ---------------------------------------------------------------------------
// Kernel 1: row-major GEMM  Y[r, n] = sum_d X[r, d] * W[n, d]
// One wave computes one 16x16 output tile via V_WMMA_F32_16X16X4_F32.
// ---------------------------------------------------------------------------
__global__ void __launch_bounds__(256)
proj_kernel(const float* __restrict__ X, const float* __restrict__ W,
            float* __restrict__ Y, int nrows, int D, int H) {
    const int wave = (blockIdx.x * blockDim.x + threadIdx.x) >> 5;
    const int lane = threadIdx.x & 31;
    const int ntiles_n = H >> 4;
    const int mt = wave / ntiles_n;          // wave-uniform
    const int nt = wave % ntiles_n;          // wave-uniform
    if (mt * 16 >= nrows) return;            // whole-wave exit: EXEC stays full

    const int l15 = lane & 15;
    const int kh  = (lane >> 4) << 1;        // K offset 0 (lanes 0-15) or 2 (16-31)

    const float* arow = X + (size_t)(mt * 16 + l15) * D + kh;
    const float* brow = W + (size_t)(nt * 16 + l15) * D + kh;

    v8f c = {0.f, 0.f, 0.f, 0.f, 0.f, 0.f, 0.f, 0.f};
    for (int kk = 0; kk < D; kk += 4) {
        v2f a = *(const v2f*)(arow + kk);
        v2f b = *(const v2f*)(brow + kk);
        c = __builtin_amdgcn_wmma_f32_16x16x4_f32(false, a, false, b,
                                                  (short)0, c, false, false);
    }

    const int mbase = mt * 16 + ((lane >> 4) << 3);
    float* yb = Y + (size_t)mbase * H + nt * 16 + l15;
#pragma unroll
    for (int r = 0; r < 8; ++r) yb[(size_t)r * H] = c[r];
}

// ---------------------------------------------------------------------------
// Tensor Data Mover: DMA a 2D fp32 tile (rows x width) from global into LDS.
// 6-arg builtin (amdgpu-toolchain signature); D# packed per CDNA5 ISA
// CDNA5 ISA.  Optional LDS padding: insert (pad_amount+1) dwords
// after every 2<<pad_interval dwords stored (bank-conflict-free k-tile).
// ---------------------------------------------------------------------------
static __device__ inline void tdm_load_2d(const void* gptr, unsigned lds_off,
                                          unsigned width, unsigned rows,
                                          unsigned stride_elems,
                                          unsigned pad_en, unsigned pad_interval,
                                          unsigned pad_amount) {
    unsigned long long ga = (unsigned long long)(uintptr_t)gptr;
    u32x4 g0 = {0u, 0u, 0u, 0u};
    g0[0] = 1u;                                   // count=1, user mode, no gather
    g0[1] = lds_off;                              // lds_addr (bytes)
    g0[2] = (unsigned)ga;                         // global_addr[31:0]
    g0[3] = (unsigned)((ga >> 32) & 0x01FFFFFFu)  // global_addr[56:32]
          | (2u << 30);                           // type = 2 ("image")
    i32x8 g1 = {0, 0, 0, 0, 0, 0, 0, 0};
    g1[0] = (int)((2u << 16)                      // data_size = 4 bytes
                | (pad_en << 20)
                | (pad_interval << 22)
                | (pad_amount << 25));
    g1[1] = (int)((width & 0xFFFFu) << 16);                       // tensor_dim0[15:0] @bits[63:48]
    g1[2] = (int)(((width >> 16) & 0xFFFFu)                       // tensor_dim0[31:16]
                | ((rows & 0xFFFFu) << 16));                      // tensor_dim1[15:0]
    g1[3] = (int)(((rows >> 16) & 0xFFFFu)                        // tensor_dim1[31:16]
                | ((width & 0xFFFFu) << 16));                     // tile_dim0 = width
    g1[4] = (int)(rows & 0xFFFFu);                                // tile_dim1 = rows (tile_dim2=0)
    g1[5] = (int)stride_elems;                                    // tensor_dim0_stride[31:0]
    i32x4 gz4 = {0, 0, 0, 0};
    i32x8 gz8 = {0, 0, 0, 0, 0, 0, 0, 0};
    __builtin_amdgcn_tensor_load_to_lds(g0, g1, gz4, gz4, gz8, 0);
}

// ---------------------------------------------------------------------------
// Kernel 2: scores[b,q,k] = sum_h w_v[h] * tanh(qp[b,q,h] + kp[b,k,h]),
// masked to -1e6 for k >= valid_lens[b].  H fixed at 128.
// Tiles staged by the Tensor Data Mover; k-tile DMA-padded to stride 132
// (16B-aligned rows -> ds_load_b128, (4k+h) mod 64 banking).
// tanh(x) = 1 - 2/(1 + exp(2x))  -> v_exp_f32 + v_rcp_f32.
// ---------------------------------------------------------------------------
#define HH  128
#define QT  16
#define KT  64
#define SKS 132   // k-tile LDS row stride in floats (128 data + 4 pad dwords)

__global__ void __launch_bounds__(256)
scores_kernel(const float* __restrict__ qp, const float* __restrict__ kp,
              const float* __restrict__ wv, const int* __restrict__ valid_lens,
              float* __restrict__ scores, int Q, int K) {
    __shared__ __align__(16) float sq[QT * HH];
    __shared__ __align__(16) float sk[KT * SKS];
    __shared__ __align__(16) float swv[HH];

    const int b  = blockIdx.x;
    const int qt = blockIdx.y;
    const int kt = blockIdx.z;
    const int tid = threadIdx.x;

    const float* qbase = qp + ((size_t)b * Q + qt * QT) * HH;
    const float* kbase = kp + ((size_t)b * K + kt * KT) * HH;

    for (int i = tid; i < HH; i += 256) swv[i] = wv[i];

    if (tid < 32) {   // wave 0 issues both tile DMAs, then drains TENSORcnt
        tdm_load_2d(qbase, (unsigned)(uintptr_t)sq, HH, QT, HH, 0u, 0u, 0u);
        tdm_load_2d(kbase, (unsigned)(uintptr_t)sk, HH, KT, HH, 1u, 6u, 3u);
        __builtin_amdgcn_s_wait_tensorcnt(0);
    }
    __syncthreads();

    const int vl = valid_lens[b];
    for (int p = tid; p < QT * KT; p += 256) {
        const int q = p >> 6;          // p / KT
        const int k = p & (KT - 1);    // p % KT
        const v4f* q4 = (const v4f*)(sq + q * HH);
        const v4f* k4 = (const v4f*)(sk + k * SKS);
        const v4f* w4 = (const v4f*)(swv);
        float s = 0.f;
#pragma unroll 4
        for (int h4 = 0; h4 < HH / 4; ++h4) {
            v4f xq = q4[h4];
            v4f xk = k4[h4];
            v4f xw = w4[h4];
#pragma unroll
            for (int j = 0; j < 4; ++j) {
                float x = xq[j] + xk[j];
                float t = 1.f - 2.f * __builtin_amdgcn_rcpf(1.f + __expf(2.f * x));
                s = fmaf(xw[j], t, s);
            }
        }
        const int kg = kt * KT + k;
        if (kg >= vl) s = -1e6f;
        scores[((size_t)b * Q + qt * QT + q) * K + kg] = s;
    }
}

// ---------------------------------------------------------------------------
// Kernel 3: in-place row softmax over K (one 256-thread block per (b,q) row)
// ---------------------------------------------------------------------------
__global__ void __launch_bounds__(256)
softmax_kernel(float* __restrict__ s, int K) {
    float* row = s + (size_t)blockIdx.x * K;
    __shared__ float red[256];
    const int tid = threadIdx.x;

    float m = -INFINITY;
    for (int k = tid; k < K; k += 256) m = fmaxf(m, row[k]);
    red[tid] = m;
    __syncthreads();
    for (int off = 128; off > 0; off >>= 1) {
        if (tid < off) red[tid] = fmaxf(red[tid], red[tid + off]);
        __syncthreads();
    }
    m = red[0];
    __syncthreads();

    float sum = 0.f;
    for (int k = tid; k < K; k += 256) {
        float e = __expf(row[k] - m);
        row[k] = e;
        sum += e;
    }
    red[tid] = sum;
    __syncthreads();
    for (int off = 128; off > 0; off >>= 1) {
        if (tid < off) red[tid] += red[tid + off];
        __syncthreads();
    }
    const float inv = __builtin_amdgcn_rcpf(red[0]);
    for (int k = tid; k < K; k += 256) row[k] *= inv;
}

// ---------------------------------------------------------------------------
// Kernel 4: O[b,q,v] = sum_k Wt[b,q,k] * V[b,k,v]  via WMMA f32 16x16x4.
// ---------------------------------------------------------------------------
__global__ void __launch_bounds__(256)
av_kernel(const float* __restrict__ Wt, const float* __restrict__ V,
          float* __restrict__ O, int B, int Q, int K, int DV) {
    const int wave = (blockIdx.x * blockDim.x + threadIdx.x) >> 5;
    const int lane = threadIdx.x & 31;
    const int vt_count = DV >> 4;
    const int qt_count = Q >> 4;
    const int b = wave / (qt_count * vt_count);     // wave-uniform
    if (b >= B) return;                              // whole-wave exit
    const int rem = wave % (qt_count * vt_count);
    const int qt  = rem / vt_count;
    const int vt  = rem % vt_count;

    const int l15 = lane & 15;
    const int kh  = (lane >> 4) << 1;

    const float* arow = Wt + ((size_t)b * Q + qt * 16 + l15) * K + kh;
    const float* vcol = V + ((size_t)b * K + kh) * DV + vt * 16 + l15;

    v8f c = {0.f, 0.f, 0.f, 0.f, 0.f, 0.f, 0.f, 0.f};
    for (int kk = 0; kk < K; kk += 4) {
        v2f a = *(const v2f*)(arow + kk);
        v2f bb;
        bb.x = vcol[(size_t)kk * DV];
        bb.y = vcol[(size_t)(kk + 1) * DV];
        c = __builtin_amdgcn_wmma_f32_16x16x4_f32(false, a, false, bb,
                                                  (short)0, c, false, false);
    }

    const int mbase = qt * 16 + ((lane >> 4) << 3);
    float* ob = O + ((size_t)b * Q + mbase) * DV + vt * 16 + l15;
#pragma unroll
    for (int r = 0; r < 8; ++r) ob[(size_t)r * DV] = c[r];
}

// ---------------------------------------------------------------------------
// Host launcher
// inputs: queries[B,Q,DQ], keys[B,K,DK], values[B,K,DV], valid_lens[B],
//         W_q[H,DQ], W_k[H,DK], w_v[H]
// ---------------------------------------------------------------------------
extern "C" void kernel_launch(void* const* d_in, const int* in_sizes, int n_in,
                              void* d_out, int out_size, void* d_ws, size_t ws_size,
                              hipStream_t stream) {
    const float* queries    = (const float*)d_in[0];
    const float* keys       = (const float*)d_in[1];
    const float* values     = (const float*)d_in[2];
    const int*   valid_lens = (const int*)  d_in[3];
    const float* W_q        = (const float*)d_in[4];
    const float* W_k        = (const float*)d_in[5];
    const float* w_v        = (const float*)d_in[6];

    const int B  = in_sizes[3];
    const int H  = in_sizes[6];              // 128
    const int DQ = in_sizes[4] / H;          // 256
    const int DK = in_sizes[5] / H;          // 256
    const int Q  = in_sizes[0] / (B * DQ);   // 256
    const int K  = in_sizes[1] / (B * DK);   // 1024
    const int DV = in_sizes[2] / (B * K);    // 128

    // Workspace layout: qp [B*Q*H] | kp [B*K*H] | scores [B*Q*K]  (~13 MB fp32)
    float* qp     = (float*)d_ws;
    float* kp     = qp + (size_t)B * Q * H;
    float* scores = kp + (size_t)B * K * H;

    {   // q projection
        int nrows = B * Q;
        int waves = (nrows / 16) * (H / 16);
        proj_kernel<<<(waves + 7) / 8, 256, 0, stream>>>(queries, W_q, qp, nrows, DQ, H);
    }
    {   // k projection
        int nrows = B * K;
        int waves = (nrows / 16) * (H / 16);
        proj_kernel<<<(waves + 7) / 8, 256, 0, stream>>>(keys, W_k, kp, nrows, DK, H);
    }

    scores_kernel<<<dim3(B, Q / QT, K / KT), 256, 0, stream>>>(
        qp, kp, w_v, valid_lens, scores, Q, K);

    softmax_kernel<<<B * Q, 256, 0, stream>>>(scores, K);

    {
        int waves = B * (Q / 16) * (DV / 16);
        av_kernel<<<(waves + 7) / 8, 256, 0, stream>>>(scores, values, (float*)d_out,
                                                       B, Q, K, DV);
    }
}